// EncoderPreTre_81363860455571
// MI455X (gfx1250) — compile-verified
//
#include <hip/hip_runtime.h>
#include <math.h>

// ---- problem constants ----
#define BATCH 8
#define SEQ   512
#define DMODEL 1024
#define HEADS 16
#define DK    64
#define NLAYER 4
#define MTOT  (BATCH*SEQ)           // 4096
#define NMAT  ((size_t)MTOT*DMODEL) // 4194304 floats per [B,S,D] tensor
#define WSTRIDE ((size_t)DMODEL*DMODEL)
#define DIAGF (1.0f - 0.999999f)

typedef __attribute__((ext_vector_type(2))) float v2f;
typedef __attribute__((ext_vector_type(8))) float v8f;

// fp32 WMMA: D(16x16) = A(16x4) * B(4x16) + C
// A lane layout: M = lane%16, a[j] = A[M][2*(lane/16)+j]  (K index = vgpr + 2*half)
// B lane layout: N = lane%16, b[j] = B[2*(lane/16)+j][N]
// C/D layout:    element (vgpr r, lane L): M = r + 8*(L/16), N = L%16
static __device__ __forceinline__ v8f wmma4(v2f a, v2f b, v8f c) {
    return __builtin_amdgcn_wmma_f32_16x16x4_f32(false, a, false, b, (short)0, c,
                                                 false, false);
}

// ---------------- embedding gather: emb = table[src] ----------------
__global__ __launch_bounds__(256) void embed_kernel(const int* __restrict__ src,
                                                    const float* __restrict__ table,
                                                    float* __restrict__ emb) {
    int m = blockIdx.x;                 // 0..4095
    int t = threadIdx.x;                // 0..255, one float4 each (1024 floats/row)
    int tok = src[m];
    const float4* srow = (const float4*)(table + (size_t)tok * DMODEL);
    ((float4*)(emb + (size_t)m * DMODEL))[t] = srow[t];
}

// ---------------- contrib = (rs - r)/(rs+1e-9)*(1-r)*100 ----------------
__global__ __launch_bounds__(256) void contrib_kernel(const float* __restrict__ reaches,
                                                      float* __restrict__ contrib) {
    __shared__ float red[256];
    int b = blockIdx.x, t = threadIdx.x;
    const float* rb = reaches + b * SEQ;
    red[t] = rb[t] + rb[t + 256];
    __syncthreads();
    for (int s = 128; s > 0; s >>= 1) {
        if (t < s) red[t] += red[t + s];
        __syncthreads();
    }
    float rs = red[0];
    for (int i = 0; i < 2; ++i) {
        int s = t + 256 * i;
        float r = rb[s];
        contrib[b * SEQ + s] = (rs - r) / (rs + 1e-9f) * (1.0f - r) * 100.0f;
    }
}

// ---------------- QKV projection: out = x @ W^T, stored [B,H,S,DK] ----------------
// grid (16, 16, 3), block 256 (8 waves). Wave tile 32(M) x 64(N): two A frags share
// the same 4 B frags -> 6 b64 loads per 8 WMMAs.
__global__ __launch_bounds__(256) void qkv_kernel(const float* __restrict__ x,
                                                  const float* __restrict__ QW,
                                                  const float* __restrict__ KW,
                                                  const float* __restrict__ VW,
                                                  float* __restrict__ Qo,
                                                  float* __restrict__ Ko,
                                                  float* __restrict__ Vo) {
    int z = blockIdx.z;
    const float* W = (z == 0) ? QW : (z == 1) ? KW : VW;
    float* dst = (z == 0) ? Qo : (z == 1) ? Ko : Vo;

    int w = threadIdx.x >> 5, lane = threadIdx.x & 31;
    int lm = lane & 15, h2 = lane >> 4;
    int m0 = blockIdx.x * 256 + w * 32;
    int n0 = blockIdx.y * 64;

    const float* arow0 = x + (size_t)(m0 + lm) * DMODEL + 2 * h2;
    const float* arow1 = arow0 + (size_t)16 * DMODEL;
    const float* brow0 = W + (size_t)(n0 + lm) * DMODEL + 2 * h2;

    v8f acc[2][4] = {};
#pragma unroll 4
    for (int k0 = 0; k0 < DMODEL; k0 += 4) {
        v2f a0 = *(const v2f*)(arow0 + k0);
        v2f a1 = *(const v2f*)(arow1 + k0);
#pragma unroll
        for (int t = 0; t < 4; ++t) {
            v2f bf = *(const v2f*)(brow0 + (size_t)(16 * t) * DMODEL + k0);
            acc[0][t] = wmma4(a0, bf, acc[0][t]);
            acc[1][t] = wmma4(a1, bf, acc[1][t]);
        }
    }
#pragma unroll
    for (int i = 0; i < 2; ++i) {
#pragma unroll
        for (int t = 0; t < 4; ++t) {
#pragma unroll
            for (int r = 0; r < 8; ++r) {
                int mm = m0 + 16 * i + r + 8 * h2;   // m = b*SEQ + s
                int nn = n0 + 16 * t + lm;           // n = h*DK + dk
                int b = mm >> 9, s = mm & 511;
                int hh = nn >> 6, dk = nn & 63;
                dst[(((size_t)(b * HEADS + hh)) * SEQ + s) * DK + dk] = acc[i][t][r];
            }
        }
    }
}

// ---------------- attention: flash-style online softmax ----------------
// grid (B*H=128, SEQ/128=4), block 256 (8 waves, 16 q rows each).
__global__ __launch_bounds__(256) void attn_kernel(const float* __restrict__ Q,
                                                   const float* __restrict__ K,
                                                   const float* __restrict__ V,
                                                   const float* __restrict__ reaches,
                                                   const float* __restrict__ contrib,
                                                   float* __restrict__ AO) {
    __shared__ float pbuf[8][16 * 18];   // per-wave P tile, padded stride 18
    int w = threadIdx.x >> 5, lane = threadIdx.x & 31;
    int lm = lane & 15, h2 = lane >> 4;
    int bh = blockIdx.x, b = bh >> 4;
    int qbase = blockIdx.y * 128 + w * 16;

    const float* Qp = Q + (size_t)bh * SEQ * DK;
    const float* Kp = K + (size_t)bh * SEQ * DK;
    const float* Vp = V + (size_t)bh * SEQ * DK;

    // preload Q A-fragments for this wave's 16 rows (DK=64 -> 16 k-steps)
    v2f aq[16];
    {
        const float* qr = Qp + (size_t)(qbase + lm) * DK + 2 * h2;
#pragma unroll
        for (int kk = 0; kk < 16; ++kk) aq[kk] = *(const v2f*)(qr + 4 * kk);
    }

    float mrun[8], lrun[8];
#pragma unroll
    for (int r = 0; r < 8; ++r) { mrun[r] = -3.4e38f; lrun[r] = 0.0f; }
    v8f acc[4] = {};
    float* pb = pbuf[w];

    for (int n0 = 0; n0 < SEQ; n0 += 16) {
        // scores tile: s[q=16][k=16] = Q * K^T  (B frag: K rows are K-contiguous)
        v8f s = {};
        const float* kr = Kp + (size_t)(n0 + lm) * DK + 2 * h2;
#pragma unroll
        for (int kk = 0; kk < 16; ++kk) {
            v2f bk = *(const v2f*)(kr + 4 * kk);
            s = wmma4(aq[kk], bk, s);
        }
        float rch = reaches[b * SEQ + n0 + lm];

        float fs[8];
#pragma unroll
        for (int r = 0; r < 8; ++r) {
            float sv = s[r] * 0.125f;            // 1/sqrt(64)
            // row max across the 16 lanes of this half-wave
            float t = sv;
            t = fmaxf(t, __shfl_xor(t, 1, 32));
            t = fmaxf(t, __shfl_xor(t, 2, 32));
            t = fmaxf(t, __shfl_xor(t, 4, 32));
            t = fmaxf(t, __shfl_xor(t, 8, 32));
            float mn = fmaxf(mrun[r], t);
            float f = expf(mrun[r] - mn);
            float p = expf(sv - mn);
            float ps = p;
            ps += __shfl_xor(ps, 1, 32);
            ps += __shfl_xor(ps, 2, 32);
            ps += __shfl_xor(ps, 4, 32);
            ps += __shfl_xor(ps, 8, 32);
            lrun[r] = lrun[r] * f + ps;
            mrun[r] = mn;
            fs[r] = f;
            // effective weight: reaches[k] and near-diagonal kill (post-softmax)
            int qrow = qbase + r + 8 * h2;
            float pe = p * rch;
            if (n0 + lm == qrow) pe *= DIAGF;
            pb[(r + 8 * h2) * 18 + lm] = pe;     // C-layout -> LDS
        }
        // rescale output accumulators (row mapping identical to fs[])
#pragma unroll
        for (int t4 = 0; t4 < 4; ++t4)
#pragma unroll
            for (int r = 0; r < 8; ++r) acc[t4][r] *= fs[r];

        // P @ V : read P back in A-layout from LDS, V B-frags strided over seq rows
#pragma unroll
        for (int kk4 = 0; kk4 < 4; ++kk4) {
            v2f ap = *(const v2f*)(pb + lm * 18 + 4 * kk4 + 2 * h2);
            int row = n0 + 4 * kk4 + 2 * h2;
#pragma unroll
            for (int t4 = 0; t4 < 4; ++t4) {
                v2f bv;
                bv[0] = Vp[(size_t)row * DK + 16 * t4 + lm];
                bv[1] = Vp[(size_t)(row + 1) * DK + 16 * t4 + lm];
                acc[t4] = wmma4(ap, bv, acc[t4]);
            }
        }
    }

    // epilogue: out = (v_q - acc/l) * contrib[q]
#pragma unroll
    for (int t4 = 0; t4 < 4; ++t4) {
        int dk = 16 * t4 + lm;
#pragma unroll
        for (int r = 0; r < 8; ++r) {
            int qrow = qbase + r + 8 * h2;
            float vv = Vp[(size_t)qrow * DK + dk];
            float o = (vv - acc[t4][r] / lrun[r]) * contrib[b * SEQ + qrow];
            AO[((size_t)bh * SEQ + qrow) * DK + dk] = o;
        }
    }
}

// ---------------- output projection + residual: xout = xin + AO @ OW^T ----------------
// AO read in [B,H,S,DK] layout (k decomposed into head/dk). grid (16,16), block 256.
// Same 32x64 wave tile as qkv_kernel.
__global__ __launch_bounds__(256) void oproj_kernel(const float* __restrict__ AO,
                                                    const float* __restrict__ OW,
                                                    const float* __restrict__ xin,
                                                    float* __restrict__ xout) {
    int w = threadIdx.x >> 5, lane = threadIdx.x & 31;
    int lm = lane & 15, h2 = lane >> 4;
    int m0 = blockIdx.x * 256 + w * 32;
    int n0 = blockIdx.y * 64;

    int mA0 = m0 + lm;
    int mA1 = m0 + 16 + lm;
    int b0 = mA0 >> 9, s0 = mA0 & 511;
    int b1 = mA1 >> 9, s1 = mA1 & 511;
    const float* abase0 = AO + ((size_t)b0 * HEADS) * SEQ * DK + (size_t)s0 * DK;
    const float* abase1 = AO + ((size_t)b1 * HEADS) * SEQ * DK + (size_t)s1 * DK;
    const float* brow0  = OW + (size_t)(n0 + lm) * DMODEL + 2 * h2;

    v8f acc[2][4] = {};
#pragma unroll 4
    for (int k0 = 0; k0 < DMODEL; k0 += 4) {
        int k = k0 + 2 * h2;                       // pairs never cross a head boundary
        size_t aoff = (size_t)(k >> 6) * (SEQ * DK) + (k & 63);
        v2f a0 = *(const v2f*)(abase0 + aoff);
        v2f a1 = *(const v2f*)(abase1 + aoff);
#pragma unroll
        for (int t = 0; t < 4; ++t) {
            v2f bf = *(const v2f*)(brow0 + (size_t)(16 * t) * DMODEL + k0);
            acc[0][t] = wmma4(a0, bf, acc[0][t]);
            acc[1][t] = wmma4(a1, bf, acc[1][t]);
        }
    }
#pragma unroll
    for (int i = 0; i < 2; ++i) {
#pragma unroll
        for (int t = 0; t < 4; ++t) {
#pragma unroll
            for (int r = 0; r < 8; ++r) {
                int mm = m0 + 16 * i + r + 8 * h2, nn = n0 + 16 * t + lm;
                size_t idx = (size_t)mm * DMODEL + nn;
                xout[idx] = xin[idx] + acc[i][t][r];
            }
        }
    }
}

extern "C" void kernel_launch(void* const* d_in, const int* in_sizes, int n_in,
                              void* d_out, int out_size, void* d_ws, size_t ws_size,
                              hipStream_t stream) {
    const int*   src     = (const int*)d_in[0];
    const float* reaches = (const float*)d_in[1];
    const float* table   = (const float*)d_in[2];
    const float* qw      = (const float*)d_in[3];
    const float* kw      = (const float*)d_in[4];
    const float* vw      = (const float*)d_in[5];
    const float* ow      = (const float*)d_in[6];

    float* out    = (float*)d_out;
    float* emb    = out;            // tuple output 0
    float* xfinal = out + NMAT;     // tuple output 1

    float* ws = (float*)d_ws;
    float* W0 = ws;                 // x ping
    float* W1 = ws + NMAT;          // x pong
    float* Qb = ws + 2 * NMAT;
    float* Kb = ws + 3 * NMAT;
    float* Vb = ws + 4 * NMAT;
    float* AO = ws + 5 * NMAT;
    float* CT = ws + 6 * NMAT;      // contrib [B,S]

    embed_kernel<<<MTOT, 256, 0, stream>>>(src, table, emb);
    contrib_kernel<<<BATCH, 256, 0, stream>>>(reaches, CT);

    for (int L = 0; L < NLAYER; ++L) {
        const float* xin = (L == 0) ? emb : (L == 1) ? W0 : (L == 2) ? W1 : W0;
        float* xout      = (L == 0) ? W0  : (L == 1) ? W1 : (L == 2) ? W0 : xfinal;

        qkv_kernel<<<dim3(MTOT / 256, DMODEL / 64, 3), 256, 0, stream>>>(
            xin, qw + (size_t)L * WSTRIDE, kw + (size_t)L * WSTRIDE,
            vw + (size_t)L * WSTRIDE, Qb, Kb, Vb);

        attn_kernel<<<dim3(BATCH * HEADS, SEQ / 128), 256, 0, stream>>>(
            Qb, Kb, Vb, reaches, CT, AO);

        oproj_kernel<<<dim3(MTOT / 256, DMODEL / 64), 256, 0, stream>>>(
            AO, ow + (size_t)L * WSTRIDE, xin, xout);
    }
}